// EDL_CAFM_74663711473884
// MI455X (gfx1250) — compile-verified
//
#include <hip/hip_runtime.h>
#include <hip/hip_bf16.h>

typedef __attribute__((ext_vector_type(2))) float v2f;
typedef __attribute__((ext_vector_type(8))) float v8f;

#define B_    4
#define C_    64
#define CR_   32
#define H_    256
#define W_    256
#define HW_   65536
#define NEG_INF (-3.402823466e38f)

// ---- workspace layout (float offsets) ----
#define OFF_AVG1 0
#define OFF_MAX1 256
#define OFF_AVG2 512
#define OFF_MAX2 768
#define OFF_S1   1024
#define OFF_S2   (1024 + 16384)
#define OFF_P1   (OFF_S2 + 16384)            // pooled1 [B][2][HW]
#define OFF_P2   (OFF_P1 + B_*2*HW_)
#define OFF_Y1A  (OFF_P2 + B_*2*HW_)         // conv1 out, tensor1 [B][HW]
#define OFF_YA   (OFF_Y1A + B_*HW_)          // conv2 out, tensor1
#define OFF_Y1B  (OFF_YA + B_*HW_)
#define OFF_YB   (OFF_Y1B + B_*HW_)
#define OFF_MX   (OFF_YB + B_*HW_)           // [2][B]
#define OFF_SM   (OFF_MX + 8)                // [2][B]

// ============================================================
// Kernel 1: per-(tensor,b,c) mean + max over HW (512 rows)
// ============================================================
__global__ void __launch_bounds__(256) rowreduce_kernel(
    const float* __restrict__ f1, const float* __restrict__ f2,
    float* __restrict__ ws)
{
  const int row = blockIdx.x;           // 0..511
  const int tensor = row >> 8;
  const int bc = row & 255;
  const int tid = threadIdx.x;
  const float4* src = (const float4*)((tensor ? f2 : f1) + (size_t)bc * HW_);
  float sum = 0.0f, mx = NEG_INF;
  for (int i = tid; i < HW_ / 4; i += 256) {
    float4 v = src[i];
    sum += (v.x + v.y) + (v.z + v.w);
    mx = fmaxf(mx, fmaxf(fmaxf(v.x, v.y), fmaxf(v.z, v.w)));
  }
  __shared__ float ssum[256];
  __shared__ float smax[256];
  ssum[tid] = sum; smax[tid] = mx;
  __syncthreads();
  for (int s = 128; s > 0; s >>= 1) {
    if (tid < s) {
      ssum[tid] += ssum[tid + s];
      smax[tid] = fmaxf(smax[tid], smax[tid + s]);
    }
    __syncthreads();
  }
  if (tid == 0) {
    float* avg = ws + (tensor ? OFF_AVG2 : OFF_AVG1);
    float* mxo = ws + (tensor ? OFF_MAX2 : OFF_MAX1);
    avg[bc] = ssum[0] * (1.0f / (float)HW_);
    mxo[bc] = smax[0];
  }
}

// ============================================================
// Kernel 2: channel MLPs, cross outer-product, row/col softmax
//           -> S1[B][64][64], S2[B][64][64]
// ============================================================
__global__ void __launch_bounds__(256) desc_kernel(
    float* __restrict__ ws,
    const float* __restrict__ w_avg1, const float* __restrict__ b_avg1,
    const float* __restrict__ w_max1, const float* __restrict__ b_max1,
    const float* __restrict__ w_avg11, const float* __restrict__ b_avg11,
    const float* __restrict__ w_max11, const float* __restrict__ b_max11,
    const float* __restrict__ w_avg2, const float* __restrict__ b_avg2,
    const float* __restrict__ w_max2, const float* __restrict__ b_max2,
    const float* __restrict__ w_avg22, const float* __restrict__ b_avg22,
    const float* __restrict__ w_max22, const float* __restrict__ b_max22)
{
  __shared__ float sd[4][256];     // avg1,max1,avg2,max2
  __shared__ float sH[4][B_][CR_]; // hidden: [a1avg,a1max,a2avg,a2max]
  __shared__ float sA1[B_][C_];
  __shared__ float sA2[B_][C_];
  const int tid = threadIdx.x;
  sd[0][tid] = ws[OFF_AVG1 + tid];
  sd[1][tid] = ws[OFF_MAX1 + tid];
  sd[2][tid] = ws[OFF_AVG2 + tid];
  sd[3][tid] = ws[OFF_MAX2 + tid];
  __syncthreads();

  for (int task = tid; task < 512; task += 256) {
    const int which = task >> 7, rem = task & 127, b = rem >> 5, r = rem & 31;
    const float *w, *bb, *in;
    switch (which) {
      case 0:  w = w_avg1; bb = b_avg1; in = &sd[0][b * 64]; break;
      case 1:  w = w_max1; bb = b_max1; in = &sd[1][b * 64]; break;
      case 2:  w = w_avg2; bb = b_avg2; in = &sd[2][b * 64]; break;
      default: w = w_max2; bb = b_max2; in = &sd[3][b * 64]; break;
    }
    float acc = bb[r];
    for (int c = 0; c < 64; ++c) acc += w[r * 64 + c] * in[c];
    sH[which][b][r] = fmaxf(acc, 0.0f);
  }
  __syncthreads();

  for (int task = tid; task < 512; task += 256) {
    const int sel = task >> 8, idx = task & 255, b = idx >> 6, c = idx & 63;
    if (sel == 0) {
      float acc = b_avg11[c] + b_max11[c];
      for (int r = 0; r < 32; ++r)
        acc += w_avg11[c * 32 + r] * sH[0][b][r] + w_max11[c * 32 + r] * sH[1][b][r];
      sA1[b][c] = acc;
    } else {
      float acc = b_avg22[c] + b_max22[c];
      for (int r = 0; r < 32; ++r)
        acc += w_avg22[c * 32 + r] * sH[2][b][r] + w_max22[c * 32 + r] * sH[3][b][r];
      sA2[b][c] = acc;
    }
  }
  __syncthreads();

  const int b = tid >> 6, i = tid & 63;
  // S1 row i: softmax_j( a1[i]*a2[j] )
  {
    const float a1i = sA1[b][i];
    float m = NEG_INF;
    for (int j = 0; j < 64; ++j) m = fmaxf(m, a1i * sA2[b][j]);
    float s = 0.0f;
    for (int j = 0; j < 64; ++j) s += expf(a1i * sA2[b][j] - m);
    const float inv = 1.0f / s;
    float* S1 = ws + OFF_S1 + (b * 64 + i) * 64;
    for (int j = 0; j < 64; ++j) S1[j] = expf(a1i * sA2[b][j] - m) * inv;
  }
  // S2 row p=i: softmax_q( a1[q]*a2[p] )
  {
    const float a2p = sA2[b][i];
    float m = NEG_INF;
    for (int q = 0; q < 64; ++q) m = fmaxf(m, sA1[b][q] * a2p);
    float s = 0.0f;
    for (int q = 0; q < 64; ++q) s += expf(sA1[b][q] * a2p - m);
    const float inv = 1.0f / s;
    float* S2 = ws + OFF_S2 + (b * 64 + i) * 64;
    for (int q = 0; q < 64; ++q) S2[q] = expf(sA1[b][q] * a2p - m) * inv;
  }
}

// ============================================================
// Kernel 3: fused (S @ f) + channel mean/max pooling  [WMMA f32]
//   at[i,n] = sum_j S[i,j] f[j,n];  pooled = mean_i/max_i
//   grid = B * (HW/128); block = 256 (8 waves x 16 columns)
// ============================================================
__global__ void __launch_bounds__(256) attn_pool_kernel(
    const float* __restrict__ f,    // [B][64][HW]
    const float* __restrict__ S,    // [B][64][64]
    float* __restrict__ pooled)     // [B][2][HW]  (mean, max)
{
  __shared__ float Slds[C_ * C_];   // 16 KB
  const int b    = blockIdx.x >> 9;       // 512 tiles per batch
  const int tile = blockIdx.x & 511;
  const int tid  = threadIdx.x;

  const float* Sg = S + b * 4096;
  for (int i = tid; i < 4096; i += 256) Slds[i] = Sg[i];
  __syncthreads();

  const int wave = tid >> 5;
  const int lane = tid & 31;
  const int lmod = lane & 15;
  const int lhi  = lane >> 4;             // 0 or 1
  const int ncol = tile * 128 + wave * 16 + lmod;

  const float* fb = f + (size_t)b * C_ * HW_;

  v8f acc0 = {}, acc1 = {}, acc2 = {}, acc3 = {};
  for (int k = 0; k < 16; ++k) {
    const int j0 = 4 * k + 2 * lhi;       // K base for this lane half
    // B fragment (4x16 f32): rows j0, j0+1 at column ncol
    v2f bfrag;
    bfrag.x = fb[(size_t)j0 * HW_ + ncol];
    bfrag.y = fb[(size_t)(j0 + 1) * HW_ + ncol];
    // A fragments (16x4 f32) from LDS, one per M-tile
    const v2f a0 = *(const v2f*)&Slds[(lmod)      * 64 + j0];
    const v2f a1 = *(const v2f*)&Slds[(16 + lmod) * 64 + j0];
    const v2f a2 = *(const v2f*)&Slds[(32 + lmod) * 64 + j0];
    const v2f a3 = *(const v2f*)&Slds[(48 + lmod) * 64 + j0];
    acc0 = __builtin_amdgcn_wmma_f32_16x16x4_f32(false, a0, false, bfrag, (short)0, acc0, false, false);
    acc1 = __builtin_amdgcn_wmma_f32_16x16x4_f32(false, a1, false, bfrag, (short)0, acc1, false, false);
    acc2 = __builtin_amdgcn_wmma_f32_16x16x4_f32(false, a2, false, bfrag, (short)0, acc2, false, false);
    acc3 = __builtin_amdgcn_wmma_f32_16x16x4_f32(false, a3, false, bfrag, (short)0, acc3, false, false);
  }

  // lane covers M = mtile*16 + r + 8*lhi for r=0..7, column = ncol
  float psum = 0.0f, pmax = NEG_INF;
  for (int r = 0; r < 8; ++r) {
    psum += ((acc0[r] + acc1[r]) + (acc2[r] + acc3[r]));
    pmax = fmaxf(pmax, fmaxf(fmaxf(acc0[r], acc1[r]), fmaxf(acc2[r], acc3[r])));
  }
  // combine the two lane halves (M rows 0-7 <-> rows 8-15 of each M-tile)
  const float osum = __shfl_xor(psum, 16, 32);
  const float omax = __shfl_xor(pmax, 16, 32);
  psum += osum;
  pmax = fmaxf(pmax, omax);
  if (lane < 16) {
    pooled[((size_t)b * 2) * HW_ + ncol]     = psum * (1.0f / (float)C_);
    pooled[((size_t)b * 2 + 1) * HW_ + ncol] = pmax;
  }
}

// ============================================================
// Kernel 4/5: 3x3 SAME convs on pooled map
// ============================================================
__global__ void __launch_bounds__(256) conv1_kernel(
    const float* __restrict__ pooled, const float* __restrict__ w,
    const float* __restrict__ bias, float* __restrict__ y1)
{
  const int idx = blockIdx.x * 256 + threadIdx.x;   // 0..B*HW
  const int b = idx >> 16, n = idx & 65535;
  const int h = n >> 8, x = n & 255;
  float acc = bias[0];
  for (int ci = 0; ci < 2; ++ci)
    for (int dy = 0; dy < 3; ++dy) {
      const int hh = h + dy - 1;
      if ((unsigned)hh >= 256u) continue;
      for (int dx = 0; dx < 3; ++dx) {
        const int xx = x + dx - 1;
        if ((unsigned)xx >= 256u) continue;
        acc += pooled[((size_t)b * 2 + ci) * HW_ + hh * 256 + xx] * w[ci * 9 + dy * 3 + dx];
      }
    }
  y1[idx] = fmaxf(acc, 0.0f);
}

__global__ void __launch_bounds__(256) conv2_kernel(
    const float* __restrict__ y1, const float* __restrict__ w,
    const float* __restrict__ bias, float* __restrict__ y)
{
  const int idx = blockIdx.x * 256 + threadIdx.x;
  const int b = idx >> 16, n = idx & 65535;
  const int h = n >> 8, x = n & 255;
  float acc = bias[0];
  for (int dy = 0; dy < 3; ++dy) {
    const int hh = h + dy - 1;
    if ((unsigned)hh >= 256u) continue;
    for (int dx = 0; dx < 3; ++dx) {
      const int xx = x + dx - 1;
      if ((unsigned)xx >= 256u) continue;
      acc += y1[(size_t)b * HW_ + hh * 256 + xx] * w[dy * 3 + dx];
    }
  }
  y[idx] = acc;
}

// ============================================================
// Kernel 6/7: per-batch spatial softmax stats over HW
// ============================================================
__global__ void __launch_bounds__(256) maxred_kernel(
    const float* __restrict__ y, float* __restrict__ mx)
{
  const int b = blockIdx.x;
  const float* yb = y + (size_t)b * HW_;
  float m = NEG_INF;
  for (int i = threadIdx.x; i < HW_; i += 256) m = fmaxf(m, yb[i]);
  __shared__ float s[256];
  s[threadIdx.x] = m;
  __syncthreads();
  for (int st = 128; st > 0; st >>= 1) {
    if (threadIdx.x < st) s[threadIdx.x] = fmaxf(s[threadIdx.x], s[threadIdx.x + st]);
    __syncthreads();
  }
  if (threadIdx.x == 0) mx[b] = s[0];
}

__global__ void __launch_bounds__(256) expsum_kernel(
    const float* __restrict__ y, const float* __restrict__ mx,
    float* __restrict__ sm)
{
  const int b = blockIdx.x;
  const float m = mx[b];
  const float* yb = y + (size_t)b * HW_;
  float acc = 0.0f;
  for (int i = threadIdx.x; i < HW_; i += 256) acc += expf(yb[i] - m);
  __shared__ float s[256];
  s[threadIdx.x] = acc;
  __syncthreads();
  for (int st = 128; st > 0; st >>= 1) {
    if (threadIdx.x < st) s[threadIdx.x] += s[threadIdx.x + st];
    __syncthreads();
  }
  if (threadIdx.x == 0) sm[b] = s[0];
}

// ============================================================
// Kernel 8: o = f * (1 + softmax(y))   (gate broadcast over C)
// ============================================================
__global__ void __launch_bounds__(256) final_kernel(
    const float* __restrict__ f, const float* __restrict__ y,
    const float* __restrict__ mx, const float* __restrict__ sm,
    float* __restrict__ o)
{
  const size_t i4 = (size_t)blockIdx.x * 256 + threadIdx.x; // over B*C*HW/4
  const size_t e = i4 * 4;
  const int b = (int)(e >> 22);          // C*HW = 2^22
  const int n = (int)(e & 65535);
  const float m = mx[b];
  const float inv = 1.0f / sm[b];
  const float4 fv = ((const float4*)f)[i4];
  const float4 yv = *(const float4*)(y + (size_t)b * HW_ + n);
  float4 ov;
  ov.x = fv.x * (1.0f + expf(yv.x - m) * inv);
  ov.y = fv.y * (1.0f + expf(yv.y - m) * inv);
  ov.z = fv.z * (1.0f + expf(yv.z - m) * inv);
  ov.w = fv.w * (1.0f + expf(yv.w - m) * inv);
  ((float4*)o)[i4] = ov;
}

// ============================================================
extern "C" void kernel_launch(void* const* d_in, const int* in_sizes, int n_in,
                              void* d_out, int out_size, void* d_ws, size_t ws_size,
                              hipStream_t stream) {
  const float* f1      = (const float*)d_in[0];
  const float* f2      = (const float*)d_in[1];
  const float* w_avg1  = (const float*)d_in[2];
  const float* b_avg1  = (const float*)d_in[3];
  const float* w_max1  = (const float*)d_in[4];
  const float* b_max1  = (const float*)d_in[5];
  const float* w_avg11 = (const float*)d_in[6];
  const float* b_avg11 = (const float*)d_in[7];
  const float* w_max11 = (const float*)d_in[8];
  const float* b_max11 = (const float*)d_in[9];
  const float* w_avg2  = (const float*)d_in[10];
  const float* b_avg2  = (const float*)d_in[11];
  const float* w_max2  = (const float*)d_in[12];
  const float* b_max2  = (const float*)d_in[13];
  const float* w_avg22 = (const float*)d_in[14];
  const float* b_avg22 = (const float*)d_in[15];
  const float* w_max22 = (const float*)d_in[16];
  const float* b_max22 = (const float*)d_in[17];
  const float* conv1_w = (const float*)d_in[18];
  const float* conv1_b = (const float*)d_in[19];
  const float* conv2_w = (const float*)d_in[20];
  const float* conv2_b = (const float*)d_in[21];

  float* ws = (float*)d_ws;
  float* o1 = (float*)d_out;
  float* o2 = o1 + (size_t)B_ * C_ * HW_;

  // 1: descriptors (mean/max over HW) for both tensors
  rowreduce_kernel<<<2 * B_ * C_, 256, 0, stream>>>(f1, f2, ws);

  // 2: MLPs + cross softmaxes -> S1, S2
  desc_kernel<<<1, 256, 0, stream>>>(ws,
      w_avg1, b_avg1, w_max1, b_max1, w_avg11, b_avg11, w_max11, b_max11,
      w_avg2, b_avg2, w_max2, b_max2, w_avg22, b_avg22, w_max22, b_max22);

  // 3: WMMA attention-apply fused with channel pooling
  attn_pool_kernel<<<B_ * (HW_ / 128), 256, 0, stream>>>(f1, ws + OFF_S1, ws + OFF_P1);
  attn_pool_kernel<<<B_ * (HW_ / 128), 256, 0, stream>>>(f2, ws + OFF_S2, ws + OFF_P2);

  // 4/5: spatial gate convs
  conv1_kernel<<<B_ * HW_ / 256, 256, 0, stream>>>(ws + OFF_P1, conv1_w, conv1_b, ws + OFF_Y1A);
  conv1_kernel<<<B_ * HW_ / 256, 256, 0, stream>>>(ws + OFF_P2, conv1_w, conv1_b, ws + OFF_Y1B);
  conv2_kernel<<<B_ * HW_ / 256, 256, 0, stream>>>(ws + OFF_Y1A, conv2_w, conv2_b, ws + OFF_YA);
  conv2_kernel<<<B_ * HW_ / 256, 256, 0, stream>>>(ws + OFF_Y1B, conv2_w, conv2_b, ws + OFF_YB);

  // 6/7: spatial softmax stats
  maxred_kernel<<<B_, 256, 0, stream>>>(ws + OFF_YA, ws + OFF_MX);
  maxred_kernel<<<B_, 256, 0, stream>>>(ws + OFF_YB, ws + OFF_MX + B_);
  expsum_kernel<<<B_, 256, 0, stream>>>(ws + OFF_YA, ws + OFF_MX, ws + OFF_SM);
  expsum_kernel<<<B_, 256, 0, stream>>>(ws + OFF_YB, ws + OFF_MX + B_, ws + OFF_SM + B_);

  // 8: residual gate
  final_kernel<<<(B_ * C_ * HW_ / 4) / 256, 256, 0, stream>>>(f1, ws + OFF_YA, ws + OFF_MX, ws + OFF_SM, o1);
  final_kernel<<<(B_ * C_ * HW_ / 4) / 256, 256, 0, stream>>>(f2, ws + OFF_YB, ws + OFF_MX + B_, ws + OFF_SM + B_, o2);
}